// VanillaGNNLayer_7017976562249
// MI455X (gfx1250) — compile-verified
//
#include <hip/hip_runtime.h>

typedef float v2f __attribute__((ext_vector_type(2)));
typedef float v8f __attribute__((ext_vector_type(8)));

#define DIM 128      // D_IN == D_OUT == 128
#define LDSW 132     // padded LDS row stride (conflict-free: 132 % 64 == 4)

// ---------------------------------------------------------------------------
// Kernel 1: h = x @ W^T  via V_WMMA_F32_16X16X4_F32 (full fp32 precision).
// Block = 256 threads = 8 waves; each wave computes a 16x128 slab of h.
// W (128x128 fp32, 64KB) staged in LDS with padded stride for bank-conflict-
// free ds_load_b64 B-fragment reads.
// ---------------------------------------------------------------------------
__global__ __launch_bounds__(256) void gnn_gemm_wmma(
    const float* __restrict__ x, const float* __restrict__ W,
    float* __restrict__ h, int n_nodes)
{
    __shared__ float Ws[DIM * LDSW];

    const int t = threadIdx.x;
    // Cooperative stage of W into LDS (row-major, padded stride).
    for (int i = t; i < DIM * DIM; i += 256) {
        const int r = i >> 7;
        const int c = i & 127;
        Ws[r * LDSW + c] = W[i];
    }
    __syncthreads();

    const int wave  = t >> 5;
    const int lane  = t & 31;
    const int lhalf = lane >> 4;   // 0: K=0,1 half ; 1: K=2,3 half
    const int l16   = lane & 15;
    const int m0    = blockIdx.x * 128 + wave * 16;

    // A-fragment row for this lane (clamped for the ragged tail; stores are
    // predicated instead so EXEC is all-1s through every WMMA).
    int arow = m0 + l16;
    if (arow >= n_nodes) arow = n_nodes - 1;
    const float* aptr = x + (size_t)arow * DIM + 2 * lhalf;

    v8f acc[8];
#pragma unroll
    for (int nt = 0; nt < 8; ++nt) acc[nt] = (v8f)0.0f;

    for (int k0 = 0; k0 < DIM; k0 += 4) {
        // A: 16x4 fp32 tile of x (lane layout per ISA 7.12.2).
        const v2f a = *(const v2f*)(aptr + k0);
#pragma unroll
        for (int nt = 0; nt < 8; ++nt) {
            // B[k][n] = W[n][k]  (h = x @ W^T): row n of W supplies column n of B.
            const float* bp = &Ws[(nt * 16 + l16) * LDSW + k0 + 2 * lhalf];
            const v2f b = *(const v2f*)bp;
            acc[nt] = __builtin_amdgcn_wmma_f32_16x16x4_f32(
                false, a, false, b, (short)0, acc[nt], false, false);
        }
    }

    // D layout: VGPR v -> M = v + 8*lhalf, N = nt*16 + l16.
#pragma unroll
    for (int nt = 0; nt < 8; ++nt) {
#pragma unroll
        for (int v = 0; v < 8; ++v) {
            const int row = m0 + v + 8 * lhalf;
            if (row < n_nodes)
                h[(size_t)row * DIM + nt * 16 + l16] = acc[nt][v];
        }
    }
}

// ---------------------------------------------------------------------------
// Kernel 2: zero the accumulator (d_out) — float4 stores.
// ---------------------------------------------------------------------------
__global__ __launch_bounds__(256) void gnn_zero(float4* __restrict__ p, int n4)
{
    const int i = blockIdx.x * 256 + threadIdx.x;
    if (i < n4) p[i] = make_float4(0.f, 0.f, 0.f, 0.f);
}

// ---------------------------------------------------------------------------
// Kernel 3: edge scatter  out[rows[e]] += vals[e] * h[cols[e]].
// One wave32 per edge: 32 lanes x float4 == 128 floats. Gather is a single
// global_load_b128 per lane; accumulation uses fp32 global atomics (no
// return -> STOREcnt). h and out are L2-resident (103MB << 192MB L2).
// ---------------------------------------------------------------------------
__global__ __launch_bounds__(256) void gnn_scatter(
    const float* __restrict__ h,
    const long long* __restrict__ rows,
    const long long* __restrict__ cols,
    const float* __restrict__ vals,
    float* __restrict__ out, int n_edges)
{
    const int e = blockIdx.x * 8 + (threadIdx.x >> 5);
    if (e >= n_edges) return;
    const int lane = threadIdx.x & 31;

    const long long r = rows[e];
    const long long c = cols[e];
    const float     v = vals[e];

    const float4 m = *((const float4*)(h + (size_t)c * DIM) + lane);
    float* op = out + (size_t)r * DIM + lane * 4;

    unsafeAtomicAdd(op + 0, v * m.x);
    unsafeAtomicAdd(op + 1, v * m.y);
    unsafeAtomicAdd(op + 2, v * m.z);
    unsafeAtomicAdd(op + 3, v * m.w);
}

// ---------------------------------------------------------------------------
// Kernel 4: in-place ReLU on d_out.
// ---------------------------------------------------------------------------
__global__ __launch_bounds__(256) void gnn_relu(float4* __restrict__ p, int n4)
{
    const int i = blockIdx.x * 256 + threadIdx.x;
    if (i < n4) {
        float4 v = p[i];
        v.x = fmaxf(v.x, 0.f);
        v.y = fmaxf(v.y, 0.f);
        v.z = fmaxf(v.z, 0.f);
        v.w = fmaxf(v.w, 0.f);
        p[i] = v;
    }
}

// ---------------------------------------------------------------------------
extern "C" void kernel_launch(void* const* d_in, const int* in_sizes, int n_in,
                              void* d_out, int out_size, void* d_ws, size_t ws_size,
                              hipStream_t stream)
{
    const float*     x    = (const float*)d_in[0];
    const long long* rows = (const long long*)d_in[1];   // int64 per reference
    const long long* cols = (const long long*)d_in[2];   // int64 per reference
    const float*     vals = (const float*)d_in[3];
    const float*     W    = (const float*)d_in[4];

    const int n_nodes = in_sizes[0] / DIM;   // 100000
    const int n_edges = in_sizes[1];         // 3200000

    float* out = (float*)d_out;
    float* h   = (float*)d_ws;               // needs n_nodes*128*4 = 51.2MB scratch

    const int n4 = (n_nodes * DIM) / 4;

    // 1) zero accumulator
    gnn_zero<<<(n4 + 255) / 256, 256, 0, stream>>>((float4*)out, n4);
    // 2) h = x @ W^T (WMMA fp32)
    gnn_gemm_wmma<<<(n_nodes + 127) / 128, 256, 0, stream>>>(x, W, h, n_nodes);
    // 3) scatter-add over edges (one wave per edge)
    gnn_scatter<<<(n_edges + 7) / 8, 256, 0, stream>>>(h, rows, cols, vals, out, n_edges);
    // 4) ReLU in place
    gnn_relu<<<(n4 + 255) / 256, 256, 0, stream>>>((float4*)out, n4);
}